// SelfCorrelationComputation6_584115552558
// MI455X (gfx1250) — compile-verified
//
#include <hip/hip_runtime.h>

// Problem constants (from reference): B=8, C=64, H=W=56, KATT=5 (reflect pad 2), KCONV=3.
#define B_    8
#define C_    64
#define H_    56
#define W_    56
#define HW_   3136          // 56*56
#define NPIX_ 25088         // B*HW
#define TILES_ 1568         // NPIX/16

typedef float v2f __attribute__((ext_vector_type(2)));
typedef float v4f __attribute__((ext_vector_type(4)));
typedef float v8f __attribute__((ext_vector_type(8)));
typedef unsigned int u32x4 __attribute__((ext_vector_type(4)));
typedef int i32x4 __attribute__((ext_vector_type(4)));
typedef int i32x8 __attribute__((ext_vector_type(8)));

#if __has_builtin(__builtin_amdgcn_tensor_load_to_lds) && \
    __has_builtin(__builtin_amdgcn_s_wait_tensorcnt)
#define HAVE_TDM 1
#else
#define HAVE_TDM 0
#endif

__device__ __forceinline__ int reflect56(int i) {
  if (i < 0) i = -i;            // numpy/jax 'reflect' (no edge repeat), pad<=2
  if (i >= 56) i = 110 - i;
  return i;
}

#if HAVE_TDM
// TDM DMA of 4096 contiguous f32 (16 KB) from global to LDS.
// D# packing per CDNA5 ISA ch.8: group0 = {count=1, lds_addr, global_addr[56:0],
// type=2}; group1 = {mask=0, data_size=4B, tensor_dim0=4096, tensor_dim1=1,
// tile_dim0=4096 (single row), dim0_stride=4096}; groups 2/3 zero (<=2D tensor).
// This toolchain's builtin is the 6-arg clang-23 form:
//   (uint32x4 g0, int32x8 g1, int32x4 g2, int32x4 g3, int32x8, int cpol)
__device__ __forceinline__ void tdm_load_16kb_f32(const float* gsrc,
                                                  unsigned lds_byte_off) {
  unsigned long long ga = (unsigned long long)gsrc;
  u32x4 g0;
  g0[0] = 1u;                                        // count=1, user descriptor
  g0[1] = lds_byte_off;                              // lds_addr (bytes)
  g0[2] = (unsigned)ga;                              // global_addr[31:0]
  g0[3] = (unsigned)((ga >> 32) & 0x1FFFFFFu)        // global_addr[56:32]
          | (2u << 30);                              // type=2 ("image")
  i32x8 g1;
  g1[0] = (int)(2u << 16);                           // data_size=2 -> 4 bytes
  g1[1] = (int)(4096u << 16);                        // tensor_dim0[15:0]=4096
  g1[2] = (int)(1u << 16);                           // dim0 hi=0, tensor_dim1=1
  g1[3] = (int)(4096u << 16);                        // tile_dim0=4096
  g1[4] = 0;                                         // tile_dim1/2 unused
  g1[5] = 4096;                                      // tensor_dim0_stride lo
  g1[6] = (int)(4096u << 16);                        // stride hi=0, dim1_stride lo
  g1[7] = 0;
  i32x4 gz4 = {0, 0, 0, 0};
  i32x8 gz8 = {0, 0, 0, 0, 0, 0, 0, 0};
  __builtin_amdgcn_tensor_load_to_lds(g0, g1, gz4, gz4, gz8, 0);
}
#endif

// ---------------------------------------------------------------------------
// Kernel 1: relative-position embedding (pixel-major) + collapsed conv weights
//   pe_t[pix*64 + d]   = wp[d,0]*loc_w + wp[d,1]*loc_h + bp[d]
//   wcomb[j*576+o*9+t] = sum_i w_dep[o,i,ky,kx] * w_fc[i,j]   (j: 0=q,1=k,2=v)
// ---------------------------------------------------------------------------
__global__ void __launch_bounds__(256) setup_kernel(
    const float* __restrict__ wp, const float* __restrict__ bp,
    const float* __restrict__ w_fc, const float* __restrict__ w_dep,
    float* __restrict__ pe_t, float* __restrict__ wcomb) {
  int tid = blockIdx.x * 256 + threadIdx.x;
  if (tid < HW_ * 64) {
    int pix = tid >> 6, d = tid & 63;
    int y = pix / W_, x = pix % W_;
    float lw = -1.0f + 2.0f * (float)x / 55.0f;   // linspace(-1,1,56) along W
    float lh = -1.0f + 2.0f * (float)y / 55.0f;   // linspace(-1,1,56) along H
    pe_t[tid] = wp[2 * d] * lw + wp[2 * d + 1] * lh + bp[d];
  }
  if (tid < 3 * 64 * 9) {
    int j = tid / 576;
    int rem = tid % 576;
    int o = rem / 9, t = rem % 9;
    int ky = t / 3, kx = t % 3;
    float s = 0.f;
#pragma unroll
    for (int i = 0; i < 9; ++i)
      s += w_dep[((o * 9 + i) * 3 + ky) * 3 + kx] * w_fc[i * 3 + j];
    wcomb[tid] = s;
  }
}

// ---------------------------------------------------------------------------
// Kernel 2: q/k/v = W(64x64) @ x(64 x 25088) + b, via V_WMMA_F32_16X16X4_F32.
// Weights staged into LDS by the Tensor Data Mover (wave 0 issues 3 DMAs,
// TENSORcnt wait, then block barrier); falls back to b128 copies if the TDM
// builtin is unavailable. One wave per 16-pixel tile; B-fragments loaded once
// and reused for 3 matrices x 4 channel blocks (12 tiles, 16 chained WMMAs
// each). Output written PIXEL-MAJOR: qt[pixel*64 + channel].
// ---------------------------------------------------------------------------
__global__ void __launch_bounds__(256) qkv_wmma_kernel(
    const float* __restrict__ x,
    const float* __restrict__ w1, const float* __restrict__ b1,
    const float* __restrict__ w2, const float* __restrict__ b2,
    const float* __restrict__ w3, const float* __restrict__ b3,
    float* __restrict__ qt, float* __restrict__ kt, float* __restrict__ vt) {
  __shared__ float smem[3 * 4096 + 3 * 64];   // w1|w2|w3 row-major, then b1|b2|b3
  int tid = threadIdx.x;

#if HAVE_TDM
  if (tid < 32) {   // wave 0 only: TDM ignores EXEC, gate at branch level
    // Generic LDS pointer is {aperture_hi : lds_offset}; low 32 bits = offset.
    unsigned lds_base = (unsigned)(unsigned long long)(&smem[0]);
    tdm_load_16kb_f32(w1, lds_base);
    tdm_load_16kb_f32(w2, lds_base + 16384u);
    tdm_load_16kb_f32(w3, lds_base + 32768u);
    __builtin_amdgcn_s_wait_tensorcnt(0);
  }
#else
  {
    v4f* d = (v4f*)smem;
    const v4f* s1 = (const v4f*)w1;
    const v4f* s2 = (const v4f*)w2;
    const v4f* s3 = (const v4f*)w3;
    for (int i = tid; i < 1024; i += 256) {
      d[i] = s1[i];
      d[1024 + i] = s2[i];
      d[2048 + i] = s3[i];
    }
  }
#endif
  // Biases: branch-separated direct pointers (no pointer-select -> no flat ops).
  if (tid >= 64 && tid < 128)       smem[3 * 4096 + (tid - 64)] = b1[tid - 64];
  else if (tid >= 128 && tid < 192) smem[3 * 4096 + 64 + (tid - 128)] = b2[tid - 128];
  else if (tid >= 192)              smem[3 * 4096 + 128 + (tid - 192)] = b3[tid - 192];
  __syncthreads();

  int lane = tid & 31;
  int half = lane >> 4;        // ISA wave32 frag layout: lanes 16..31 carry K+2 / M+8
  int n = lane & 15;
  int tile = blockIdx.x * 8 + (tid >> 5);      // 196 blocks * 8 waves = 1568 tiles
  int gp = tile * 16 + n;                      // global pixel (same batch across tile)
  int bb = gp / HW_;
  int p = gp % HW_;
  const float* xb = x + (size_t)bb * 64 * HW_ + p;

  // B fragments (K x N = 4 x 16), element k = kb*4 + half*2 + j per ISA layout.
  v2f bfrag[16];
#pragma unroll
  for (int kb = 0; kb < 16; ++kb) {
    int k = kb * 4 + half * 2;
    bfrag[kb].x = xb[(size_t)k * HW_];
    bfrag[kb].y = xb[(size_t)(k + 1) * HW_];
  }

  float* const outs[3] = {qt, kt, vt};
#pragma unroll
  for (int mat = 0; mat < 3; ++mat) {
    const float* wl = smem + mat * 4096;
    const float* bl = smem + 3 * 4096 + mat * 64;
#pragma unroll
    for (int mo = 0; mo < 4; ++mo) {
      const float* wrow = wl + (mo * 16 + n) * 64;   // A row m = mo*16 + (lane&15)
      v8f acc = {};
#pragma unroll
      for (int kb = 0; kb < 16; ++kb) {
        v2f a;
        a.x = wrow[kb * 4 + half * 2];
        a.y = wrow[kb * 4 + half * 2 + 1];
        // D = A(16x4) * B(4x16) + C, f32 exact
        acc = __builtin_amdgcn_wmma_f32_16x16x4_f32(
            false, a, false, bfrag[kb], (short)0, acc, false, false);
      }
      // C/D layout: lane -> n=lane&15, vgpr r -> m = r + half*8 (+ mo*16)
      int mbase = mo * 16 + half * 8;
      float* dst = outs[mat] + (size_t)gp * 64 + mbase;
      v4f o0 = {acc[0] + bl[mbase + 0], acc[1] + bl[mbase + 1],
                acc[2] + bl[mbase + 2], acc[3] + bl[mbase + 3]};
      v4f o1 = {acc[4] + bl[mbase + 4], acc[5] + bl[mbase + 5],
                acc[6] + bl[mbase + 6], acc[7] + bl[mbase + 7]};
      *(v4f*)dst = o0;
      *(v4f*)(dst + 4) = o1;
    }
  }
}

// ---------------------------------------------------------------------------
// Kernel 3: 5x5 windowed attention with reflect padding + rel-pos term.
// One wave per pixel; lane holds channels d and d+32; online softmax over the
// 25 positions (flash style), cross-lane dot via __shfl_xor tree.
// Writes rate1 * out_att to d_out in reference (B,C,H,W) layout.
// ---------------------------------------------------------------------------
__global__ void __launch_bounds__(256) attn_kernel(
    const float* __restrict__ qt, const float* __restrict__ kt,
    const float* __restrict__ vt, const float* __restrict__ pe_t,
    const float* __restrict__ rate1, float* __restrict__ out) {
  int wid = (blockIdx.x * 256 + threadIdx.x) >> 5;   // pixel id, 0..25087
  int lane = threadIdx.x & 31;
  int bb = wid / HW_;
  int p = wid % HW_;
  int y = p / W_, x = p % W_;

  const float* qv = qt + (size_t)wid * 64;
  float q0 = qv[lane] * 0.125f;          // scaling = 64^-0.5
  float q1 = qv[lane + 32] * 0.125f;
  const float* pec = pe_t + (size_t)p * 64;
  float pc0 = pec[lane], pc1 = pec[lane + 32];

  float m = -3.4e38f, s = 0.f, a0 = 0.f, a1 = 0.f;
#pragma unroll
  for (int py = 0; py < 5; ++py) {
    int yy = reflect56(y + py - 2);
#pragma unroll
    for (int px = 0; px < 5; ++px) {
      int xx = reflect56(x + px - 2);
      int np = yy * W_ + xx;
      size_t ngp = (size_t)(bb * HW_ + np) * 64;
      const float* kv = kt + ngp;
      const float* pv = pe_t + (size_t)np * 64;
      float t = q0 * (kv[lane] + pc0 - pv[lane]) +
                q1 * (kv[lane + 32] + pc1 - pv[lane + 32]);
#pragma unroll
      for (int off = 16; off > 0; off >>= 1) t += __shfl_xor(t, off, 32);
      const float* vv = vt + ngp;
      float v0 = vv[lane], v1 = vv[lane + 32];
      float mn = fmaxf(m, t);
      float c = __expf(m - mn);
      float e = __expf(t - mn);
      s = s * c + e;
      a0 = a0 * c + e * v0;
      a1 = a1 * c + e * v1;
      m = mn;
    }
  }
  float scale = rate1[0] / s;
  size_t o0 = (size_t)(bb * 64 + lane) * HW_ + p;
  out[o0] = a0 * scale;
  out[o0 + (size_t)32 * HW_] = a1 * scale;
}

// ---------------------------------------------------------------------------
// Kernel 4: collapsed conv branch. out += rate2 * (b_dep + sum_taps
//            wq*q + wk*k + wv*v), zero padding, per-channel 3x3 kernels.
// ---------------------------------------------------------------------------
__global__ void __launch_bounds__(256) conv_kernel(
    const float* __restrict__ qt, const float* __restrict__ kt,
    const float* __restrict__ vt, const float* __restrict__ wcomb,
    const float* __restrict__ b_dep, const float* __restrict__ rate2,
    float* __restrict__ out) {
  __shared__ float sw[3 * 576];
  for (int i = threadIdx.x; i < 3 * 576; i += 256) sw[i] = wcomb[i];
  __syncthreads();

  int wid = (blockIdx.x * 256 + threadIdx.x) >> 5;
  int lane = threadIdx.x & 31;
  int bb = wid / HW_, p = wid % HW_;
  int y = p / W_, x = p % W_;
  int d0 = lane, d1 = lane + 32;
  float acc0 = b_dep[d0], acc1 = b_dep[d1];
#pragma unroll
  for (int t = 0; t < 9; ++t) {
    int yy = y + t / 3 - 1, xx = x + t % 3 - 1;
    if ((unsigned)yy < 56u && (unsigned)xx < 56u) {
      size_t ngp = (size_t)(bb * HW_ + yy * W_ + xx) * 64;
      float qq0 = qt[ngp + d0], qq1 = qt[ngp + d1];
      float kk0 = kt[ngp + d0], kk1 = kt[ngp + d1];
      float vv0 = vt[ngp + d0], vv1 = vt[ngp + d1];
      acc0 += sw[d0 * 9 + t] * qq0 + sw[576 + d0 * 9 + t] * kk0 +
              sw[1152 + d0 * 9 + t] * vv0;
      acc1 += sw[d1 * 9 + t] * qq1 + sw[576 + d1 * 9 + t] * kk1 +
              sw[1152 + d1 * 9 + t] * vv1;
    }
  }
  float r2 = rate2[0];
  size_t o0 = (size_t)(bb * 64 + d0) * HW_ + p;
  out[o0] += r2 * acc0;
  out[o0 + (size_t)32 * HW_] += r2 * acc1;
}

// ---------------------------------------------------------------------------
extern "C" void kernel_launch(void* const* d_in, const int* in_sizes, int n_in,
                              void* d_out, int out_size, void* d_ws, size_t ws_size,
                              hipStream_t stream) {
  (void)in_sizes; (void)n_in; (void)out_size; (void)ws_size;
  const float* x     = (const float*)d_in[0];
  const float* w1    = (const float*)d_in[1];
  const float* b1    = (const float*)d_in[2];
  const float* w2    = (const float*)d_in[3];
  const float* b2    = (const float*)d_in[4];
  const float* w3    = (const float*)d_in[5];
  const float* b3    = (const float*)d_in[6];
  const float* wp    = (const float*)d_in[7];
  const float* bp    = (const float*)d_in[8];
  const float* w_fc  = (const float*)d_in[9];
  const float* w_dep = (const float*)d_in[10];
  const float* b_dep = (const float*)d_in[11];
  const float* rate1 = (const float*)d_in[12];
  const float* rate2 = (const float*)d_in[13];
  float* out = (float*)d_out;

  // Workspace layout (floats): qt | kt | vt (pixel-major) | pe_t | wcomb
  float* ws = (float*)d_ws;
  float* qt    = ws;
  float* kt    = qt + (size_t)NPIX_ * 64;
  float* vt    = kt + (size_t)NPIX_ * 64;
  float* pe_t  = vt + (size_t)NPIX_ * 64;
  float* wcomb = pe_t + (size_t)HW_ * 64;   // total ~19.2 MiB

  setup_kernel<<<784, 256, 0, stream>>>(wp, bp, w_fc, w_dep, pe_t, wcomb);
  qkv_wmma_kernel<<<196, 256, 0, stream>>>(x, w1, b1, w2, b2, w3, b3, qt, kt, vt);
  attn_kernel<<<3136, 256, 0, stream>>>(qt, kt, vt, pe_t, rate1, out);
  conv_kernel<<<3136, 256, 0, stream>>>(qt, kt, vt, wcomb, b_dep, rate2, out);
}